// GCN_39917426049646
// MI455X (gfx1250) — compile-verified
//
#include <hip/hip_runtime.h>
#include <hip/hip_bf16.h>
#include <math.h>

typedef float v2f __attribute__((ext_vector_type(2)));
typedef float v8f __attribute__((ext_vector_type(8)));
typedef int   v4i __attribute__((ext_vector_type(4)));

#define NN 100000   // nodes
#define FIN 512     // input features
#define HD 256      // hidden
#define CO 40       // classes
#define CP 48       // padded classes (3 x 16 WMMA col tiles)
#define KB 64       // gemm1 K chunk (16 rows x 64 k = 256 x b128)

#if __has_builtin(__builtin_amdgcn_global_load_async_to_lds_b128)
#define HAS_ASYNC_LDS 1
#else
#define HAS_ASYNC_LDS 0
#endif

__device__ __forceinline__ void async_cp_b128(const float* g, float* l) {
#if HAS_ASYNC_LDS
  __builtin_amdgcn_global_load_async_to_lds_b128(
      (__attribute__((address_space(1))) v4i*)g,
      (__attribute__((address_space(3))) v4i*)l, 0, 0);
#else
  const float4 v = *(const float4*)g;
  ((float4*)l)[0] = v;
#endif
}

__device__ __forceinline__ void wait_async() {
#if HAS_ASYNC_LDS
  asm volatile("s_wait_asynccnt 0x0" ::: "memory");
#endif
}

// ---------------------------------------------------------------------------
// GEMM1: C[NN,HD] = A[NN,FIN] @ B[FIN,HD]   (fp32 WMMA 16x16x4)
// block = 256 thr = 8 waves; block tile = 16 rows x 256 cols (full HD),
// so x streams from HBM exactly once.  A tiles (16x64) double-buffered in
// LDS via async global->LDS DMA, prefetch overlapped with 32 WMMAs/chunk.
// Each wave: two 16x16 accumulators at cols wave*16 and 128+wave*16.
// ---------------------------------------------------------------------------
__global__ __launch_bounds__(256) void gemm1_wmma(
    const float* __restrict__ A, const float* __restrict__ B,
    float* __restrict__ C) {
  __shared__ float lA[2][16][KB + 4];  // pad 4: rows land on distinct banks
  const int tid  = threadIdx.x;
  const int wave = tid >> 5, lane = tid & 31;
  const int m  = lane & 15;
  const int hi = (lane >> 4) & 1;          // lanes 16..31: K+2/K+3, M+8
  const int row0 = blockIdx.x << 4;
  const int col0 = wave << 4;

  // staging role: 16 rows x 16 quads, one b128 per thread per chunk
  const int sr = tid >> 4, sq = (tid & 15) << 2;
  const float* gsrc = A + (size_t)(row0 + sr) * FIN + sq;

  async_cp_b128(gsrc, &lA[0][sr][sq]);   // prefetch chunk 0

  v8f acc0 = {0.f, 0.f, 0.f, 0.f, 0.f, 0.f, 0.f, 0.f};
  v8f acc1 = acc0;
  int p = 0;

  for (int k0 = 0; k0 < FIN; k0 += KB) {
    wait_async();          // this wave's DMA into lA[p] landed
    __syncthreads();       // all waves' data visible; prior reads of lA[p^1] done
    if (k0 + KB < FIN)
      async_cp_b128(gsrc + k0 + KB, &lA[p ^ 1][sr][sq]);  // overlap next chunk
#pragma unroll 8
    for (int kk = 0; kk < KB; kk += 4) {
      const int ko = kk + (hi << 1);
      v2f a, b0, b1;
      a.x = lA[p][m][ko];
      a.y = lA[p][m][ko + 1];
      const size_t kr0 = (size_t)(k0 + ko) * HD;
      b0.x = B[kr0 + col0 + m];
      b0.y = B[kr0 + HD + col0 + m];
      b1.x = B[kr0 + 128 + col0 + m];
      b1.y = B[kr0 + HD + 128 + col0 + m];
      acc0 = __builtin_amdgcn_wmma_f32_16x16x4_f32(
          false, a, false, b0, (short)0, acc0, false, false);
      acc1 = __builtin_amdgcn_wmma_f32_16x16x4_f32(
          false, a, false, b1, (short)0, acc1, false, false);
    }
    p ^= 1;
  }
#pragma unroll
  for (int i = 0; i < 8; ++i) {
    const size_t r = (size_t)(row0 + i + (hi << 3)) * HD;
    C[r + col0 + m]       = acc0[i];
    C[r + 128 + col0 + m] = acc1[i];
  }
}

// ---------------------------------------------------------------------------
// GEMM2: T[NN,CP] = relu(Ag[NN,HD] + b1) @ W2pad[HD,CP]
// W2 (zero-padded to 48 cols) + b1 live in LDS; bias+ReLU fused into A load.
// block = 8 waves, each wave: 16 rows x 48 cols (3 WMMA col tiles).
// ---------------------------------------------------------------------------
__global__ __launch_bounds__(256) void gemm2_wmma(
    const float* __restrict__ Ag, const float* __restrict__ bias1,
    const float* __restrict__ W2, float* __restrict__ T) {
  __shared__ float lB[HD][CP];    // 48 KB (of 320 KB/WGP)
  __shared__ float lbias[HD];
  const int tid = threadIdx.x;
  for (int i = tid; i < HD * CP; i += 256) {
    const int k = i / CP, c = i % CP;
    lB[k][c] = (c < CO) ? W2[(size_t)k * CO + c] : 0.f;
  }
  lbias[tid] = bias1[tid];
  __syncthreads();

  const int wave = tid >> 5, lane = tid & 31;
  const int m  = lane & 15;
  const int hi = (lane >> 4) & 1;
  const int rowbase = (blockIdx.x << 7) + (wave << 4);
  if (rowbase >= NN) return;  // wave-uniform guard; 100000 % 16 == 0

  v8f acc0 = {0.f,0.f,0.f,0.f,0.f,0.f,0.f,0.f};
  v8f acc1 = acc0, acc2 = acc0;
  const int row = rowbase + m;

  for (int k0 = 0; k0 < HD; k0 += 4) {
    const int ko = k0 + (hi << 1);
    const float2 av = *(const float2*)(Ag + (size_t)row * HD + ko);
    v2f a, b0, b1v, b2v;
    a.x = fmaxf(av.x + lbias[ko], 0.f);
    a.y = fmaxf(av.y + lbias[ko + 1], 0.f);
    b0.x  = lB[ko][m];       b0.y  = lB[ko + 1][m];
    b1v.x = lB[ko][16 + m];  b1v.y = lB[ko + 1][16 + m];
    b2v.x = lB[ko][32 + m];  b2v.y = lB[ko + 1][32 + m];
    acc0 = __builtin_amdgcn_wmma_f32_16x16x4_f32(false, a, false, b0,
                                                 (short)0, acc0, false, false);
    acc1 = __builtin_amdgcn_wmma_f32_16x16x4_f32(false, a, false, b1v,
                                                 (short)0, acc1, false, false);
    acc2 = __builtin_amdgcn_wmma_f32_16x16x4_f32(false, a, false, b2v,
                                                 (short)0, acc2, false, false);
  }
#pragma unroll
  for (int i = 0; i < 8; ++i) {
    const size_t r = (size_t)(rowbase + i + (hi << 3)) * CP;
    T[r + m]      = acc0[i];
    T[r + 16 + m] = acc1[i];
    T[r + 32 + m] = acc2[i];
  }
}

// ---------------------------------------------------------------------------
// SpMM, 256 features: out[dst] += val * feat[src].  One wave32 per edge,
// float4 gathers (feat/out each 102 MB -> L2-resident on 192 MB L2),
// global_atomic_add_f32 scatters.
// ---------------------------------------------------------------------------
__global__ __launch_bounds__(256) void spmm256(
    const int* __restrict__ src, const int* __restrict__ dst,
    const float* __restrict__ val, const float* __restrict__ feat,
    float* __restrict__ out, int E) {
  const int lane = threadIdx.x & 31;
  const int nw = (gridDim.x * 256) >> 5;
  for (int e = (blockIdx.x * 256 + threadIdx.x) >> 5; e < E; e += nw) {
    const int s = src[e], d = dst[e];
    const float v = val[e];
    const float4* fr = (const float4*)(feat + (size_t)s * HD);
    float* orow = out + (size_t)d * HD;
#pragma unroll
    for (int j = 0; j < 2; ++j) {
      const int c = lane + (j << 5);         // float4 index, 64 per row
      const float4 f = fr[c];
      atomicAdd(orow + (c << 2) + 0, v * f.x);
      atomicAdd(orow + (c << 2) + 1, v * f.y);
      atomicAdd(orow + (c << 2) + 2, v * f.z);
      atomicAdd(orow + (c << 2) + 3, v * f.w);
    }
  }
}

// ---------------------------------------------------------------------------
// SpMM, 40 features (source stride CP=48): one wave32 per edge.
// ---------------------------------------------------------------------------
__global__ __launch_bounds__(256) void spmm40(
    const int* __restrict__ src, const int* __restrict__ dst,
    const float* __restrict__ val, const float* __restrict__ T,
    float* __restrict__ out, int E) {
  const int lane = threadIdx.x & 31;
  const int nw = (gridDim.x * 256) >> 5;
  for (int e = (blockIdx.x * 256 + threadIdx.x) >> 5; e < E; e += nw) {
    const int s = src[e], d = dst[e];
    const float v = val[e];
    const float* fr = T + (size_t)s * CP;
    float* orow = out + (size_t)d * CO;
    atomicAdd(orow + lane, v * fr[lane]);
    if (lane < CO - 32)
      atomicAdd(orow + 32 + lane, v * fr[32 + lane]);
  }
}

// ---------------------------------------------------------------------------
// log_softmax over 40 classes, +b2 fused. One wave32 per row, shfl reductions.
// ---------------------------------------------------------------------------
__global__ __launch_bounds__(256) void logsoftmax40(
    const float* __restrict__ Z, const float* __restrict__ b2,
    float* __restrict__ out) {
  const int lane = threadIdx.x & 31;
  const int row = (blockIdx.x * 256 + threadIdx.x) >> 5;
  if (row >= NN) return;
  const float* z = Z + (size_t)row * CO;
  const float z0 = z[lane] + b2[lane];
  const float z1 = (lane < 8) ? (z[32 + lane] + b2[32 + lane]) : -INFINITY;
  float mx = fmaxf(z0, z1);
#pragma unroll
  for (int o = 16; o; o >>= 1) mx = fmaxf(mx, __shfl_xor(mx, o, 32));
  float s = __expf(z0 - mx) + ((lane < 8) ? __expf(z1 - mx) : 0.f);
#pragma unroll
  for (int o = 16; o; o >>= 1) s += __shfl_xor(s, o, 32);
  const float lse = mx + __logf(s);
  out[(size_t)row * CO + lane] = z0 - lse;
  if (lane < 8) out[(size_t)row * CO + 32 + lane] = z1 - lse;
}

// ---------------------------------------------------------------------------
extern "C" void kernel_launch(void* const* d_in, const int* in_sizes, int n_in,
                              void* d_out, int out_size, void* d_ws,
                              size_t ws_size, hipStream_t stream) {
  const float* x    = (const float*)d_in[0];
  const int*   esrc = (const int*)d_in[1];
  const int*   edst = (const int*)d_in[2];
  const float* eval = (const float*)d_in[3];
  const float* W1   = (const float*)d_in[4];
  const float* b1   = (const float*)d_in[5];
  const float* W2   = (const float*)d_in[6];
  const float* b2   = (const float*)d_in[7];
  const int E = in_sizes[1];

  float* r0 = (float*)d_ws;              // h0 [NN,HD], later reused as T [NN,CP]
  float* r1 = r0 + (size_t)NN * HD;      // agg1 [NN,HD], later reused as agg2 [NN,CO]
  float* out = (float*)d_out;

  // agg1 = 0
  (void)hipMemsetAsync(r1, 0, (size_t)NN * HD * sizeof(float), stream);
  // h0 = x @ W1  (x streamed from HBM once; 6250 row-tiles)
  gemm1_wmma<<<dim3(NN / 16), 256, 0, stream>>>(x, W1, r0);
  // agg1[dst] += val * h0[src]
  spmm256<<<dim3((E + 7) / 8), 256, 0, stream>>>(esrc, edst, eval, r0, r1, E);
  // T = relu(agg1 + b1) @ W2   (into r0, stride CP)
  gemm2_wmma<<<dim3((NN + 127) / 128), 256, 0, stream>>>(r1, b1, W2, r0);
  // agg2 = 0 (r1 reuse; stream order guarantees gemm2 finished reading r1)
  (void)hipMemsetAsync(r1, 0, (size_t)NN * CO * sizeof(float), stream);
  // agg2[dst] += val * T[src]
  spmm40<<<dim3((E + 7) / 8), 256, 0, stream>>>(esrc, edst, eval, r0, r1, E);
  // out = log_softmax(agg2 + b2)
  logsoftmax40<<<dim3((NN + 7) / 8), 256, 0, stream>>>(r1, b2, out);
}